// NNBase_14061722927363
// MI455X (gfx1250) — compile-verified
//
#include <hip/hip_runtime.h>

#define T_STEPS 128
#define N_B     64
#define D_IN    512
#define H_DIM   512
#define G3      (3 * H_DIM)   // 1536
#define NWG     16
#define HPAD    (H_DIM + 4)   // 516: +4 floats/row -> conflict-free LDS b64 reads

typedef float v2f __attribute__((ext_vector_type(2)));
typedef float v8f __attribute__((ext_vector_type(8)));

__device__ __forceinline__ v8f wmma_f32(v2f a, v2f b, v8f c) {
  // D = A(16x4 f32) * B(4x16 f32) + C(16x16 f32)
  return __builtin_amdgcn_wmma_f32_16x16x4_f32(
      /*neg_a=*/false, a, /*neg_b=*/false, b,
      /*c_mod=*/(short)0, c, /*reuse_a=*/false, /*reuse_b=*/false);
}

__device__ __forceinline__ float sigmoidf_(float v) {
  return 1.0f / (1.0f + __expf(-v));
}

// ---------------------------------------------------------------------------
// Kernel 1: gi = x @ w_ih^T + b_ih   ([8192,512] x [512,1536])
// Each wave computes a 32x32 output tile; block of 8 waves covers 64x128.
// ---------------------------------------------------------------------------
__global__ __launch_bounds__(256) void gi_gemm(const float* __restrict__ x,
                                               const float* __restrict__ w_ih,
                                               const float* __restrict__ b_ih,
                                               float* __restrict__ gi) {
  const int lane = threadIdx.x & 31;
  const int wave = threadIdx.x >> 5;
  const int lh   = lane & 15;
  const int kk   = (lane >> 4) << 1;          // 0 or 2: K sub-offset per half-wave
  const int M0   = blockIdx.x * 64 + (wave & 1) * 32;
  const int G0   = blockIdx.y * 128 + (wave >> 1) * 32;

  v8f c00 = {}, c01 = {}, c10 = {}, c11 = {};
  const float* xa = x    + (M0 + lh) * D_IN + kk;
  const float* wb = w_ih + (G0 + lh) * D_IN + kk;   // B^T rows are w_ih rows

  for (int k = 0; k < D_IN; k += 4) {
    v2f a0 = *(const v2f*)(xa + k);
    v2f a1 = *(const v2f*)(xa + 16 * D_IN + k);
    v2f b0 = *(const v2f*)(wb + k);
    v2f b1 = *(const v2f*)(wb + 16 * D_IN + k);
    c00 = wmma_f32(a0, b0, c00);
    c01 = wmma_f32(a0, b1, c01);
    c10 = wmma_f32(a1, b0, c10);
    c11 = wmma_f32(a1, b1, c11);
  }

  const int rowHalf = (lane >> 4) * 8;
  const int col     = lane & 15;
  v8f acc[2][2] = {{c00, c01}, {c10, c11}};
  for (int i = 0; i < 2; ++i)
    for (int j = 0; j < 2; ++j) {
      const int gcol = G0 + j * 16 + col;
      const float bias = b_ih[gcol];
      for (int q = 0; q < 8; ++q) {
        const int row = M0 + i * 16 + rowHalf + q;
        gi[row * G3 + gcol] = acc[i][j][q] + bias;
      }
    }
}

// ---------------------------------------------------------------------------
// Kernel 2: persistent masked-GRU scan. 16 WGs, WG w owns h columns
// [w*32, w*32+32). Device-scope sense barrier between steps.
// ---------------------------------------------------------------------------
__global__ __launch_bounds__(256) void gru_scan(const float* __restrict__ hxs,
                                                const float* __restrict__ masks,
                                                const float* __restrict__ w_hh,
                                                const float* __restrict__ b_hh,
                                                const float* __restrict__ gi,
                                                float* __restrict__ h_buf,
                                                unsigned* __restrict__ bar,
                                                float* __restrict__ out) {
  extern __shared__ float hlds[];             // [N_B][HPAD] masked h, 132 KB
  const int tid  = threadIdx.x;
  const int wg   = blockIdx.x;                // 0..15
  const int lane = tid & 31;
  const int wave = tid >> 5;
  const int lh   = lane & 15;
  const int kk   = (lane >> 4) << 1;
  const int m0   = (wave & 3) * 16;           // row strip
  const int j0   = wg * 32 + (wave >> 2) * 16;// 16-col h tile this wave owns

  const float* wr = w_hh + (j0 + lh) * H_DIM + kk;
  const float* wz = w_hh + (H_DIM + j0 + lh) * H_DIM + kk;
  const float* wn = w_hh + (2 * H_DIM + j0 + lh) * H_DIM + kk;
  const int rowHalf = (lane >> 4) * 8;
  const int col     = lane & 15;
  const float br_ = b_hh[j0 + col];
  const float bz_ = b_hh[H_DIM + j0 + col];
  const float bn_ = b_hh[2 * H_DIM + j0 + col];

  for (int t = 0; t < T_STEPS; ++t) {
    if (t > 0) {
      // device-scope grid barrier (sense = generation counter == t)
      __threadfence();
      __syncthreads();
      if (tid == 0) {
        unsigned ticket = __hip_atomic_fetch_add(&bar[0], 1u, __ATOMIC_ACQ_REL,
                                                 __HIP_MEMORY_SCOPE_AGENT);
        if (ticket == NWG - 1) {
          __hip_atomic_store(&bar[0], 0u, __ATOMIC_RELAXED,
                             __HIP_MEMORY_SCOPE_AGENT);
          __hip_atomic_fetch_add(&bar[1], 1u, __ATOMIC_RELEASE,
                                 __HIP_MEMORY_SCOPE_AGENT);
        } else {
          while (__hip_atomic_load(&bar[1], __ATOMIC_ACQUIRE,
                                   __HIP_MEMORY_SCOPE_AGENT) < (unsigned)t)
            __builtin_amdgcn_s_sleep(1);
        }
      }
      __syncthreads();
      __threadfence();                        // acquire: invalidate WGP$
    }

    // Stage masked h into LDS: h = h_prev * mask[t]
    const float* hsrc = (t == 0) ? hxs : h_buf;
    for (int idx = tid; idx < N_B * H_DIM / 4; idx += 256) {
      const int n  = idx >> 7;                // 128 float4 per row
      const int k4 = (idx & 127) << 2;
      const float m = masks[t * N_B + n];
      float4 v = *(const float4*)(hsrc + n * H_DIM + k4);
      float* d = hlds + n * HPAD + k4;
      d[0] = v.x * m; d[1] = v.y * m; d[2] = v.z * m; d[3] = v.w * m;
    }

    // Prefetch this step's gi cachelines (consumed after the K-loop; each
    // 16-float gate row is one 64B line). Lowers to global_prefetch_b8.
    {
      const float* gibase = gi + (size_t)(t * N_B + m0 + rowHalf) * G3;
      for (int q = 0; q < 8; ++q) {
        const float* girow = gibase + (size_t)q * G3;
        __builtin_prefetch(girow + j0 + col, 0, 1);
        __builtin_prefetch(girow + H_DIM + j0 + col, 0, 1);
        __builtin_prefetch(girow + 2 * H_DIM + j0 + col, 0, 1);
      }
    }
    __syncthreads();

    // gh tiles for the three gates of this wave's 16x16 (M x j) block
    v8f cr = {}, cz = {}, cn = {};
    const float* ha = hlds + (m0 + lh) * HPAD + kk;
    for (int k = 0; k < H_DIM; k += 4) {
      v2f a  = *(const v2f*)(ha + k);         // shared A frag (ds_load_b64)
      v2f vr = *(const v2f*)(wr + k);
      v2f vz = *(const v2f*)(wz + k);
      v2f vn = *(const v2f*)(wn + k);
      cr = wmma_f32(a, vr, cr);
      cz = wmma_f32(a, vz, cz);
      cn = wmma_f32(a, vn, cn);
    }

    // gates + state update
    for (int q = 0; q < 8; ++q) {
      const int row = m0 + rowHalf + q;
      const float* girow = gi + (size_t)(t * N_B + row) * G3;
      const float ir  = girow[j0 + col];
      const float iz  = girow[H_DIM + j0 + col];
      const float in_ = girow[2 * H_DIM + j0 + col];
      const float rg = sigmoidf_(ir + cr[q] + br_);
      const float zg = sigmoidf_(iz + cz[q] + bz_);
      const float nc = tanhf(in_ + rg * (cn[q] + bn_));
      const float hp = hlds[row * HPAD + j0 + col];
      const float hn = (1.0f - zg) * nc + zg * hp;
      out[(size_t)(t * N_B + row) * H_DIM + j0 + col] = hn;
      if (t == T_STEPS - 1)
        out[(size_t)T_STEPS * N_B * H_DIM + row * H_DIM + j0 + col] = hn;
      else
        h_buf[row * H_DIM + j0 + col] = hn;
    }
  }
}

__global__ void copy_misc(const float* __restrict__ src, float* __restrict__ dst) {
  const int i = blockIdx.x * 256 + threadIdx.x;
  dst[i] = src[i];
}

__global__ void init_bar(unsigned* bar) {
  if (threadIdx.x < 2) bar[threadIdx.x] = 0u;
}

extern "C" void kernel_launch(void* const* d_in, const int* in_sizes, int n_in,
                              void* d_out, int out_size, void* d_ws, size_t ws_size,
                              hipStream_t stream) {
  const float* x      = (const float*)d_in[0];
  const float* hxs    = (const float*)d_in[1];
  const float* hxs_1  = (const float*)d_in[2];
  const float* masks  = (const float*)d_in[3];
  const float* w_ih   = (const float*)d_in[4];
  const float* w_hh   = (const float*)d_in[5];
  const float* b_ih   = (const float*)d_in[6];
  const float* b_hh   = (const float*)d_in[7];
  float* out = (float*)d_out;

  char* ws = (char*)d_ws;
  float*    gi    = (float*)ws;                                      // 50.3 MB
  float*    h_buf = (float*)(ws + (size_t)T_STEPS * N_B * G3 * 4);   // 128 KB
  unsigned* bar   = (unsigned*)(ws + (size_t)T_STEPS * N_B * G3 * 4
                                   + (size_t)N_B * H_DIM * 4);

  const int ldsBytes = N_B * HPAD * (int)sizeof(float);              // 132 KB
  // Opt in to >default dynamic LDS (CDNA5 WGP has 320 KB). Host-side,
  // deterministic, graph-capture-safe (not a stream/alloc operation).
  (void)hipFuncSetAttribute((const void*)gru_scan,
                            hipFuncAttributeMaxDynamicSharedMemorySize,
                            ldsBytes);

  init_bar<<<1, 32, 0, stream>>>(bar);
  gi_gemm<<<dim3((T_STEPS * N_B) / 64, G3 / 128), 256, 0, stream>>>(x, w_ih, b_ih, gi);
  copy_misc<<<(N_B * H_DIM) / 256, 256, 0, stream>>>(
      hxs_1, out + (size_t)(T_STEPS * N_B + N_B) * H_DIM);
  gru_scan<<<NWG, 256, ldsBytes, stream>>>(
      hxs, masks, w_hh, b_hh, gi, h_buf, bar, out);
}